// EfficientLinearAttention_40870908789489
// MI455X (gfx1250) — compile-verified
//
#include <hip/hip_runtime.h>
#include <hip/hip_bf16.h>
#include <math.h>

typedef __attribute__((ext_vector_type(2))) float v2f;
typedef __attribute__((ext_vector_type(8))) float v8f;

#define NH 8
#define HD 64
#define NTOK 4096
#define CCH 512
#define NCHUNK 8

__device__ __forceinline__ v8f wmma4(v2f a, v2f b, v8f c) {
  // D = A(16x4, f32) x B(4x16, f32) + C(16x16, f32)  -> v_wmma_f32_16x16x4_f32
  return __builtin_amdgcn_wmma_f32_16x16x4_f32(false, a, false, b, (short)0, c,
                                               false, false);
}

__device__ __forceinline__ float elu_eps(float y) {
  return ((y > 0.f) ? y : expm1f(y)) + 1e-6f;
}

// lane^16 swap as a VALU op (v_permlanex16_b32) -- keeps the DS pipe free for
// the WMMA B-fragment loads instead of ds_bpermute.
__device__ __forceinline__ float swap16(float x) {
  const int i = __float_as_int(x);
  return __int_as_float(__builtin_amdgcn_permlanex16(
      i, i, 0x76543210u, 0xFEDCBA98u, false, false));
}
// within-half xor shuffle via v_permlane16_b32 (sel tables encode lane^mask)
__device__ __forceinline__ float perm16(float x, unsigned s1, unsigned s2) {
  const int i = __float_as_int(x);
  return __int_as_float(__builtin_amdgcn_permlane16(i, i, s1, s2, false, false));
}

// flat address of a __shared__ object: low 32 bits == LDS byte offset
__device__ __forceinline__ unsigned lds_lo32(const void* p) {
  return (unsigned)(unsigned long long)p;
}

// CDNA5 async global->LDS copy (tracked with ASYNCcnt, no VGPR round-trip)
__device__ __forceinline__ void async_b128(unsigned lds, const void* g) {
  asm volatile("global_load_async_to_lds_b128 %0, %1, off"
               :: "v"(lds), "v"((unsigned long long)g) : "memory");
}
__device__ __forceinline__ void wait_async0() {
  asm volatile("s_wait_asynccnt 0x0" ::: "memory");
}

// ---------------------------------------------------------------------------
// Kernel 1: per (b,h,chunk of 512 tokens):
//   kf = elu(K_tok @ W^T + b) + eps  (WMMA feature map)
//   partial kv   += kf^T @ V         (WMMA outer product, K=4 tokens/instr)
//   partial ksum += sum_n kf
// ---------------------------------------------------------------------------
__global__ __launch_bounds__(256) void fla_kv_kernel(
    const float* __restrict__ K, const float* __restrict__ V,
    const float* __restrict__ fmw, const float* __restrict__ fmb,
    float* __restrict__ kvp, float* __restrict__ ksump) {
  __shared__ float s_fmw[64 * 68];    // fm_w[e][d], row stride 68 (pad)
  __shared__ float s_v[8][64 * 20];   // per-wave V tile [e][tok], stride 20
  __shared__ float s_ksum[64];

  const int tid = threadIdx.x;
  const int w   = tid >> 5;
  const int L   = tid & 31;
  const int lo  = L & 15;
  const int hi  = L >> 4;

  // stage fm_w (padded rows) via async copies; barrier publishes to all waves
  {
    const unsigned lb = lds_lo32(&s_fmw[0]);
    #pragma unroll
    for (int it = 0; it < 4; ++it) {
      const int c  = tid + it * 256;   // 1024 chunks of 4 floats
      const int e  = c >> 4;
      const int d4 = (c & 15) * 4;
      async_b128(lb + (unsigned)(e * 68 + d4) * 4u, fmw + c * 4);
    }
    wait_async0();
  }
  __syncthreads();

  const int blk   = blockIdx.x;            // = bh*NCHUNK + chunk
  const int chunk = blk & (NCHUNK - 1);
  const int bh    = blk >> 3;
  const int h     = bh & (NH - 1);
  const int b     = bh >> 3;

  const float* kbase = K + ((size_t)(b * CCH + h * HD)) * NTOK;
  const float* vbase = V + ((size_t)(b * CCH + h * HD)) * NTOK;

  v8f acc[4][4];  // kv accumulators: [d-tile][e-tile]
  #pragma unroll
  for (int i = 0; i < 4; ++i)
    #pragma unroll
    for (int j = 0; j < 4; ++j)
      #pragma unroll
      for (int r = 0; r < 8; ++r) acc[i][j][r] = 0.f;

  float ksacc[4] = {0.f, 0.f, 0.f, 0.f};

  const unsigned vlb = lds_lo32(&s_v[w][0]);

  for (int t = 0; t < 4; ++t) {        // 4 x 16-token tiles per wave
    const int tb = chunk * 512 + (w * 4 + t) * 16;

    // stage V tile (16 tok x 64 e) into wave-private LDS asynchronously;
    // no block barrier needed -- region is wave-private.
    #pragma unroll
    for (int i = 0; i < 8; ++i) {
      const int e  = i * 8 + (L >> 2);
      const int tk = (L & 3) * 4;
      async_b128(vlb + (unsigned)(e * 20 + tk) * 4u,
                 vbase + (size_t)e * NTOK + tb + tk);
    }
    wait_async0();

    // A-fragments of K tokens: m=token=lo, k=d=4kc+2hi+{0,1}
    v2f ka[16];
    #pragma unroll
    for (int kc = 0; kc < 16; ++kc) {
      const int d = kc * 4 + hi * 2;
      const float* p = kbase + (size_t)d * NTOK + tb + lo;
      ka[kc].x = p[0];
      ka[kc].y = p[NTOK];
    }

    // Feature-map GEMM: kf[token][e] = sum_d K[token][d] * W[e][d]
    v8f kf[4];
    #pragma unroll
    for (int et = 0; et < 4; ++et) {
      v8f c;
      #pragma unroll
      for (int r = 0; r < 8; ++r) c[r] = 0.f;
      const int e = et * 16 + lo;
      #pragma unroll
      for (int kc = 0; kc < 16; ++kc) {
        const int d0 = kc * 4 + hi * 2;
        v2f bw;
        bw.x = s_fmw[e * 68 + d0];
        bw.y = s_fmw[e * 68 + d0 + 1];
        c = wmma4(ka[kc], bw, c);
      }
      const float bias = fmb[e];
      float s = 0.f;
      #pragma unroll
      for (int r = 0; r < 8; ++r) {
        const float y = elu_eps(c[r] + bias);
        c[r] = y;
        s += y;
      }
      kf[et] = c;
      ksacc[et] += s;
    }

    // kv GEMM: kv[d][e] += sum_tok kf[tok][d] * V[tok][e]
    // A frags from kf D-frags: feature already sits in lane&15; only the
    // token halves need the cross-half swap (VALU permlanex16).
    #pragma unroll
    for (int dt = 0; dt < 4; ++dt) {
      float dv[8], sv[8];
      #pragma unroll
      for (int r = 0; r < 8; ++r) {
        dv[r] = kf[dt][r];
        sv[r] = swap16(kf[dt][r]);
      }
      const bool lw = (hi == 0);
      v2f A[4];
      A[0].x = lw ? dv[0] : sv[2];  A[0].y = lw ? dv[1] : sv[3];
      A[1].x = lw ? dv[4] : sv[6];  A[1].y = lw ? dv[5] : sv[7];
      A[2].x = lw ? sv[0] : dv[2];  A[2].y = lw ? sv[1] : dv[3];
      A[3].x = lw ? sv[4] : dv[6];  A[3].y = lw ? sv[5] : dv[7];
      #pragma unroll
      for (int et = 0; et < 4; ++et) {
        const int e = et * 16 + lo;
        #pragma unroll
        for (int tc = 0; tc < 4; ++tc) {
          v2f bv;
          const int tk = tc * 4 + hi * 2;
          bv.x = s_v[w][e * 20 + tk];
          bv.y = s_v[w][e * 20 + tk + 1];
          acc[dt][et] = wmma4(A[tc], bv, acc[dt][et]);
        }
      }
    }
  }

  // fold token halves of ksum (same e on both halves)
  float kh[4];
  #pragma unroll
  for (int et = 0; et < 4; ++et)
    kh[et] = ksacc[et] + swap16(ksacc[et]);

  // deterministic cross-wave reduction (barrier-serialized, no fp atomics)
  float* s_red = &s_v[0][0];  // reuse: 4096 floats needed, 10240 available
  for (int wv = 0; wv < 8; ++wv) {
    __syncthreads();
    if (w == wv) {
      #pragma unroll
      for (int dt = 0; dt < 4; ++dt)
        #pragma unroll
        for (int et = 0; et < 4; ++et)
          #pragma unroll
          for (int r = 0; r < 8; ++r) {
            const int d = dt * 16 + r + hi * 8;
            const int e = et * 16 + lo;
            const float x = acc[dt][et][r];
            if (wv == 0) s_red[d * 64 + e] = x;
            else         s_red[d * 64 + e] += x;
          }
      if (hi == 0) {
        #pragma unroll
        for (int et = 0; et < 4; ++et) {
          const int e = et * 16 + lo;
          if (wv == 0) s_ksum[e] = kh[et];
          else         s_ksum[e] += kh[et];
        }
      }
    }
  }
  __syncthreads();

  float* kvout = kvp + (size_t)blk * 4096;
  for (int i = tid; i < 4096; i += 256) kvout[i] = s_red[i];
  if (tid < 64) ksump[blk * 64 + tid] = s_ksum[tid];
}

// ---------------------------------------------------------------------------
// Kernel 2: reduce NCHUNK partials -> final kv (scaled by hd^-0.5) and ksum
// ---------------------------------------------------------------------------
__global__ __launch_bounds__(256) void fla_reduce_kernel(
    const float* __restrict__ kvp, const float* __restrict__ ksump,
    float* __restrict__ kvf, float* __restrict__ ksumf) {
  const int bh = blockIdx.x;
  for (int i = threadIdx.x; i < 4096; i += 256) {
    float s = 0.f;
    for (int c = 0; c < NCHUNK; ++c)
      s += kvp[((size_t)bh * NCHUNK + c) * 4096 + i];
    kvf[(size_t)bh * 4096 + i] = s * 0.125f;  // scale = 64^-0.5
  }
  if (threadIdx.x < 64) {
    float s = 0.f;
    for (int c = 0; c < NCHUNK; ++c)
      s += ksump[(bh * NCHUNK + c) * 64 + threadIdx.x];
    ksumf[bh * 64 + threadIdx.x] = s;
  }
}

// ---------------------------------------------------------------------------
// Kernel 3: per (b, 16-token tile): each wave owns one head.
//   qf (WMMA FM) -> normalizer (permlane tree) -> out = qf@kv / norm (WMMA,
//   qf transposed via padded LDS tile) -> fused LayerNorm over C=512 -> store
// ---------------------------------------------------------------------------
__global__ __launch_bounds__(256) void fla_out_kernel(
    const float* __restrict__ Q,
    const float* __restrict__ fmw, const float* __restrict__ fmb,
    const float* __restrict__ lnw, const float* __restrict__ lnb,
    const float* __restrict__ kvf, const float* __restrict__ ksumf,
    float* __restrict__ Out) {
  __shared__ float s_fmw[64 * 68];
  __shared__ float s_tile[16 * 516];   // [token][channel], stride 516 (pad)
  __shared__ float s_part[16 * 16 * 2];
  __shared__ float s_mu[16];
  __shared__ float s_rs[16];

  const int tid = threadIdx.x;
  const int w   = tid >> 5;
  const int L   = tid & 31;
  const int lo  = L & 15;
  const int hi  = L >> 4;

  {
    const unsigned lb = lds_lo32(&s_fmw[0]);
    #pragma unroll
    for (int it = 0; it < 4; ++it) {
      const int c  = tid + it * 256;
      const int e  = c >> 4;
      const int d4 = (c & 15) * 4;
      async_b128(lb + (unsigned)(e * 68 + d4) * 4u, fmw + c * 4);
    }
    wait_async0();
  }
  __syncthreads();

  const int b  = blockIdx.x >> 8;          // 256 tiles per batch
  const int tb = (blockIdx.x & 255) << 4;  // token base
  const int h  = w;                        // wave == head

  const float* qbase = Q + ((size_t)(b * CCH + h * HD)) * NTOK;

  v2f qa[16];
  #pragma unroll
  for (int kc = 0; kc < 16; ++kc) {
    const int d = kc * 4 + hi * 2;
    const float* p = qbase + (size_t)d * NTOK + tb + lo;
    qa[kc].x = p[0];
    qa[kc].y = p[NTOK];
  }

  v8f qf[4];
  #pragma unroll
  for (int et = 0; et < 4; ++et) {
    v8f c;
    #pragma unroll
    for (int r = 0; r < 8; ++r) c[r] = 0.f;
    const int e = et * 16 + lo;
    #pragma unroll
    for (int kc = 0; kc < 16; ++kc) {
      const int d0 = kc * 4 + hi * 2;
      v2f bw;
      bw.x = s_fmw[e * 68 + d0];
      bw.y = s_fmw[e * 68 + d0 + 1];
      c = wmma4(qa[kc], bw, c);
    }
    const float bias = fmb[e];
    #pragma unroll
    for (int r = 0; r < 8; ++r) c[r] = elu_eps(c[r] + bias);
    qf[et] = c;
  }

  // normalizer = qf . ksum + eps; VALU permlane16 xor-tree over the 16 lanes
  float inv[8];
  {
    float ks[4];
    #pragma unroll
    for (int et = 0; et < 4; ++et) ks[et] = ksumf[h * 64 + et * 16 + lo];
    #pragma unroll
    for (int r = 0; r < 8; ++r) {
      float s = 0.f;
      #pragma unroll
      for (int et = 0; et < 4; ++et) s += qf[et][r] * ks[et];
      s += perm16(s, 0x67452301u, 0xEFCDAB89u);  // xor 1
      s += perm16(s, 0x54761032u, 0xDCFE98BAu);  // xor 2
      s += perm16(s, 0x32107654u, 0xBA98FEDCu);  // xor 4
      s += perm16(s, 0xFEDCBA98u, 0x76543210u);  // xor 8
      inv[r] = 1.f / (s + 1e-6f);
    }
  }

  // stage qf token-major into the (wave-private) channel block of s_tile;
  // same-wave DS ops are in order, no block barrier needed here
  #pragma unroll
  for (int et = 0; et < 4; ++et)
    #pragma unroll
    for (int r = 0; r < 8; ++r)
      s_tile[(r + hi * 8) * 516 + h * 64 + et * 16 + lo] = qf[et][r];

  // transposed re-read as A-fragments: m=token=lo, k=d
  v2f qa2[16];
  #pragma unroll
  for (int kc = 0; kc < 16; ++kc) {
    const int d0 = kc * 4 + hi * 2;
    const float* p = &s_tile[lo * 516 + h * 64 + d0];
    qa2[kc].x = p[0];
    qa2[kc].y = p[1];
  }

  // out = (qf @ kv) * inv  (kv is L2-resident, 16 KB per head)
  const float* kvb = kvf + (size_t)(b * NH + h) * 4096;
  #pragma unroll
  for (int et = 0; et < 4; ++et) {
    v8f c;
    #pragma unroll
    for (int r = 0; r < 8; ++r) c[r] = 0.f;
    const int e = et * 16 + lo;
    #pragma unroll
    for (int kc = 0; kc < 16; ++kc) {
      const int d0 = kc * 4 + hi * 2;
      v2f bv;
      bv.x = kvb[d0 * 64 + e];
      bv.y = kvb[(d0 + 1) * 64 + e];
      c = wmma4(qa2[kc], bv, c);
    }
    #pragma unroll
    for (int r = 0; r < 8; ++r)
      s_tile[(r + hi * 8) * 516 + h * 64 + et * 16 + lo] = c[r] * inv[r];
  }
  __syncthreads();   // publish all heads' channel blocks for LayerNorm

  // fused LayerNorm over C=512 per token (16 threads per token)
  const int token = tid & 15;
  const int ig    = tid >> 4;   // 0..15, 32 channels each
  float sm = 0.f, sq = 0.f;
  #pragma unroll
  for (int j = 0; j < 32; ++j) {
    const float x = s_tile[token * 516 + ig * 32 + j];
    sm += x;
    sq += x * x;
  }
  s_part[(token * 16 + ig) * 2 + 0] = sm;
  s_part[(token * 16 + ig) * 2 + 1] = sq;
  __syncthreads();
  if (tid < 16) {
    float S = 0.f, SQ = 0.f;
    for (int i = 0; i < 16; ++i) {
      S  += s_part[(tid * 16 + i) * 2 + 0];
      SQ += s_part[(tid * 16 + i) * 2 + 1];
    }
    const float mu = S * (1.f / 512.f);
    float var = SQ * (1.f / 512.f) - mu * mu;
    var = fmaxf(var, 0.f);
    s_mu[tid] = mu;
    s_rs[tid] = rsqrtf(var + 1e-5f);
  }
  __syncthreads();

  const float mu = s_mu[token];
  const float rs = s_rs[token];
  float* obase = Out + (size_t)b * CCH * NTOK + tb + token;
  #pragma unroll
  for (int j = 0; j < 32; ++j) {
    const int cch = ig * 32 + j;
    const float x = s_tile[token * 516 + cch];
    obase[(size_t)cch * NTOK] = (x - mu) * rs * lnw[cch] + lnb[cch];
  }
}

// ---------------------------------------------------------------------------
extern "C" void kernel_launch(void* const* d_in, const int* in_sizes, int n_in,
                              void* d_out, int out_size, void* d_ws, size_t ws_size,
                              hipStream_t stream) {
  (void)in_sizes; (void)n_in; (void)out_size; (void)ws_size;
  const float* q   = (const float*)d_in[0];
  const float* k   = (const float*)d_in[1];
  const float* v   = (const float*)d_in[2];
  const float* fmw = (const float*)d_in[3];
  const float* fmb = (const float*)d_in[4];
  const float* lnw = (const float*)d_in[5];
  const float* lnb = (const float*)d_in[6];
  float* out = (float*)d_out;

  // workspace layout (floats): ~9.6 MB total
  float* ws    = (float*)d_ws;
  float* kvp   = ws;                         // 512 * 4096
  float* ksump = kvp + (size_t)512 * 4096;   // 512 * 64
  float* kvf   = ksump + (size_t)512 * 64;   // 64 * 4096
  float* ksumf = kvf + (size_t)64 * 4096;    // 64 * 64

  fla_kv_kernel<<<dim3(512), dim3(256), 0, stream>>>(k, v, fmw, fmb, kvp, ksump);
  fla_reduce_kernel<<<dim3(64), dim3(256), 0, stream>>>(kvp, ksump, kvf, ksumf);
  fla_out_kernel<<<dim3(2048), dim3(256), 0, stream>>>(q, fmw, fmb, lnw, lnb,
                                                       kvf, ksumf, out);
}